// RobustVectorPool2d_40527311405334
// MI455X (gfx1250) — compile-verified
//
#include <hip/hip_runtime.h>

#define BATCH 64
#define CH    256
#define KDIM  1024
#define NITER 100

typedef __attribute__((ext_vector_type(2))) float v2f;
typedef __attribute__((ext_vector_type(8))) float v8f;

static __device__ __forceinline__ v8f wmma4(v2f a, v2f b, v8f c) {
  return __builtin_amdgcn_wmma_f32_16x16x4_f32(false, a, false, b, (short)0, c, false, false);
}

// ---- coalesced transpose: xT[b,k,c] = x[b,c,k] (makes phase-2 WMMA feeds coalesced) ----
__global__ __launch_bounds__(256) void rvp_transpose(const float* __restrict__ x,
                                                     float* __restrict__ xT) {
  __shared__ float tile[32][33];
  const int b  = blockIdx.z;
  const int k0 = blockIdx.x * 32;
  const int c0 = blockIdx.y * 32;
  const float* xb = x + (size_t)b * CH * KDIM;
  float* xTb = xT + (size_t)b * CH * KDIM;
  const int tx = threadIdx.x, ty = threadIdx.y;
  #pragma unroll
  for (int j = 0; j < 32; j += 8)
    tile[ty + j][tx] = xb[(size_t)(c0 + ty + j) * KDIM + (k0 + tx)];
  __syncthreads();
  #pragma unroll
  for (int j = 0; j < 32; j += 8)
    xTb[(size_t)(k0 + ty + j) * CH + (c0 + tx)] = tile[tx][ty + j];
}

// ---- one workgroup per batch; full 100-iteration IRLS loop in-kernel, x L2-resident ----
__global__ __launch_bounds__(256) void rvp_irls(const float* __restrict__ x,
                                                const float* __restrict__ xT,
                                                float* __restrict__ out) {
  __shared__ float ly[CH];
  __shared__ float ls[KDIM];
  __shared__ float lw[KDIM];
  __shared__ float lnum[CH];
  __shared__ float lpart[8];
  __shared__ float lden;
  __shared__ float lyy;

  const int b    = blockIdx.x;
  const int tid  = threadIdx.x;
  const int wid  = tid >> 5;
  const int lane = tid & 31;
  const int laneN = lane & 15;   // N index within a 16-wide tile
  const int laneH = lane >> 4;   // half-wave -> K offset group
  // A-operand row-0 mask: only lanes 0 and 16 carry the vector (M=0 row).
  const float amask = (laneN == 0) ? 1.0f : 0.0f;

  const float* xb  = x  + (size_t)b * CH * KDIM;
  const float* xTb = xT + (size_t)b * CH * KDIM;

  // setup: s[k] = sum_c x[c,k]^2   (coalesced over k)
  {
    float s0 = 0.f, s1 = 0.f, s2 = 0.f, s3 = 0.f;
    for (int c = 0; c < CH; ++c) {
      const float* row = xb + (size_t)c * KDIM + tid;
      float a0 = row[0], a1 = row[256], a2 = row[512], a3 = row[768];
      s0 += a0 * a0; s1 += a1 * a1; s2 += a2 * a2; s3 += a3 * a3;
    }
    ls[tid] = s0; ls[tid + 256] = s1; ls[tid + 512] = s2; ls[tid + 768] = s3;
  }
  // setup: y0[c] = mean_k x[c,k]   (coalesced over c via xT), plus yy = ||y||^2
  {
    float a = 0.f;
    for (int k = 0; k < KDIM; ++k) a += xTb[(size_t)k * CH + tid];
    float yv = a * (1.0f / (float)KDIM);
    ly[tid] = yv;
    float p = yv * yv;
    #pragma unroll
    for (int off = 16; off > 0; off >>= 1) p += __shfl_xor(p, off, 32);
    if (lane == 0) lpart[wid] = p;
  }
  __syncthreads();
  if (tid == 0) {
    float t = 0.f;
    #pragma unroll
    for (int i = 0; i < 8; ++i) t += lpart[i];
    lyy = t;
  }
  __syncthreads();

  for (int it = 0; it < NITER; ++it) {
    const float yy = lyy;

    // ---- phase 1 (WMMA): t[k] = sum_c y[c]*x[c,k] over this wave's 128 k's ----
    // c-chunk outer (A loaded once per 8 WMMAs), 8 independent accumulator chains.
    const int kbase = wid * 128;
    v8f t8[8];
    #pragma unroll
    for (int j = 0; j < 8; ++j) t8[j] = (v8f){};
    for (int c0 = 0; c0 < CH; c0 += 4) {
      const int cA = c0 + (laneH << 1);
      const v2f yv = *(const v2f*)(ly + cA);          // unconditional ds_load_b64 (broadcast)
      v2f av; av.x = yv.x * amask; av.y = yv.y * amask;
      const float* pB = xb + (size_t)cA * KDIM + (kbase + laneN);
      #pragma unroll
      for (int j = 0; j < 8; ++j) {                   // B[kk,n] = x[c0+kk, k0+n]
        v2f bv;
        bv.x = pB[j * 16];
        bv.y = pB[j * 16 + KDIM];
        t8[j] = wmma4(av, bv, t8[j]);
      }
    }
    // w[k] = 1/sqrt(1 + yy - 2 t + s[k]);  D row 0 lives in acc[0], lanes 0..15
    float den_acc = 0.f;
    if (lane < 16) {
      #pragma unroll
      for (int j = 0; j < 8; ++j) {
        const int k = kbase + j * 16 + lane;
        const float t  = t8[j][0];
        const float z2 = yy - 2.0f * t + ls[k];
        const float wv = 1.0f / sqrtf(1.0f + z2);
        lw[k] = wv;
        den_acc += wv;
      }
    }
    #pragma unroll
    for (int off = 16; off > 0; off >>= 1) den_acc += __shfl_xor(den_acc, off, 32);
    if (lane == 0) lpart[wid] = den_acc;
    __syncthreads();
    if (tid == 0) {
      float t = 0.f;
      #pragma unroll
      for (int i = 0; i < 8; ++i) t += lpart[i];
      lden = t;
    }

    // ---- phase 2 (WMMA): num[c] = sum_k w[k]*xT[k,c]; wave owns c-tiles [c0, c0+16) x2 ----
    // 4 independent accumulator chains (2 c-tiles x 2 k-streams).
    {
      const int c0 = wid * 32;
      v8f n0a = {}, n1a = {}, n0b = {}, n1b = {};
      #pragma unroll 2
      for (int k0 = 0; k0 < KDIM; k0 += 8) {
        const int kA0 = k0 + (laneH << 1);
        const int kA1 = kA0 + 4;
        const v2f w0 = *(const v2f*)(lw + kA0);
        const v2f w1 = *(const v2f*)(lw + kA1);
        v2f a0; a0.x = w0.x * amask; a0.y = w0.y * amask;
        v2f a1; a1.x = w1.x * amask; a1.y = w1.y * amask;
        const float* q0 = xTb + (size_t)kA0 * CH + laneN;
        const float* q1 = xTb + (size_t)kA1 * CH + laneN;
        v2f b00, b01, b10, b11;                       // B[kk,n] = xT[k0+kk, c0+n]
        b00.x = q0[c0];          b00.y = q0[c0 + CH];
        b01.x = q0[c0 + 16];     b01.y = q0[c0 + 16 + CH];
        b10.x = q1[c0];          b10.y = q1[c0 + CH];
        b11.x = q1[c0 + 16];     b11.y = q1[c0 + 16 + CH];
        n0a = wmma4(a0, b00, n0a);
        n1a = wmma4(a0, b01, n1a);
        n0b = wmma4(a1, b10, n0b);
        n1b = wmma4(a1, b11, n1b);
      }
      if (lane < 16) {
        lnum[c0 + lane]      = n0a[0] + n0b[0];
        lnum[c0 + 16 + lane] = n1a[0] + n1b[0];
      }
    }
    __syncthreads();

    // ---- update: y = num/den; recompute yy ----
    {
      const float yv = lnum[tid] / lden;
      ly[tid] = yv;
      float p = yv * yv;
      #pragma unroll
      for (int off = 16; off > 0; off >>= 1) p += __shfl_xor(p, off, 32);
      if (lane == 0) lpart[wid] = p;
    }
    __syncthreads();
    if (tid == 0) {
      float t = 0.f;
      #pragma unroll
      for (int i = 0; i < 8; ++i) t += lpart[i];
      lyy = t;
    }
    __syncthreads();
  }

  out[(size_t)b * CH + tid] = ly[tid];
}

extern "C" void kernel_launch(void* const* d_in, const int* in_sizes, int n_in,
                              void* d_out, int out_size, void* d_ws, size_t ws_size,
                              hipStream_t stream) {
  (void)in_sizes; (void)n_in; (void)out_size; (void)ws_size;
  const float* x = (const float*)d_in[0];
  float* out = (float*)d_out;
  float* xT  = (float*)d_ws;   // 64*256*1024 floats = 64 MB scratch

  rvp_transpose<<<dim3(KDIM / 32, CH / 32, BATCH), dim3(32, 8), 0, stream>>>(x, xT);
  rvp_irls<<<dim3(BATCH), dim3(256), 0, stream>>>(x, xT, out);
}